// MentionPrunerSpanBertHoi_16131897163800
// MI455X (gfx1250) — compile-verified
//
#include <hip/hip_runtime.h>

// ---------------------------------------------------------------------------
// MentionPruner for MI455X (gfx1250, wave32, WMMA).
// Phase 1 (HBM-bound, ~1 GiB read): fused scorer using V_WMMA_F32_16X16X4_F32,
//   with the X stream moved HBM->LDS via GLOBAL_LOAD_ASYNC_TO_LDS_B128
//   (ASYNCcnt) — no VGPR staging for the hot stream.
// Phase 2: deterministic 3-pass radix-select top-k on monotone float keys.
// Phase 3: prefix-scan emit (indices come out sorted ascending) + row gather.
// ---------------------------------------------------------------------------

typedef __attribute__((ext_vector_type(2))) float v2f;
typedef __attribute__((ext_vector_type(8))) float v8f;
typedef __attribute__((ext_vector_type(4))) int   v4i;

typedef __attribute__((address_space(1))) v4i* gv4i_ptr; // global int4*
typedef __attribute__((address_space(3))) v4i* lv4i_ptr; // LDS int4*

#define D_SPAN 1024
#define HID    128
#define KC      64   // K chunk staged in LDS
#define XS_STRIDE 68 // 64 + 4 pad: A-frag reads hit 64 distinct banks
#define WT_STRIDE 68 // B-frag reads hit 64 distinct banks
#define ROWS_PER_BLOCK 128

#if defined(__has_builtin)
#if __has_builtin(__builtin_amdgcn_global_load_async_to_lds_b128)
#define HAS_ASYNC_LDS 1
#endif
#if __has_builtin(__builtin_amdgcn_s_wait_asynccnt)
#define HAS_WAIT_ASYNC_BUILTIN 1
#endif
#endif

// workspace layout (byte offsets into d_ws)
#define WS_WIDTH   0        // 32 floats: width prior scores
#define WS_SCAL    256      // 16 u32 radix-select state
#define WS_HIST1   512      // 2048 u32
#define WS_HIST2   (512 + 8192)
#define WS_HIST3   (512 + 16384) // 1024 u32
#define WS_BS      32768    // 256*2 u32 block (gt,eq) counts
#define WS_BOFF    49152    // 256*2 u32 exclusive-scanned offsets
#define WS_SELIDX  65536    // selected indices (int), k <= 3900
#define WS_KEYS    131072   // N u32 monotone keys

__device__ __forceinline__ unsigned mono_key(float f) {
  unsigned u = __float_as_uint(f);
  return (u & 0x80000000u) ? ~u : (u | 0x80000000u); // order-preserving
}

__global__ void zero_kernel(unsigned* p, int n) {
  for (int i = blockIdx.x * blockDim.x + threadIdx.x; i < n;
       i += gridDim.x * blockDim.x) p[i] = 0u;
}

// ---- tiny width-prior MLP: wscore[w] = relu(emb[w]@Ww1+bw1)@Ww2 + bw2 -----
__global__ void width_score_kernel(const float* __restrict__ emb,
                                   const float* __restrict__ Ww1,
                                   const float* __restrict__ bw1,
                                   const float* __restrict__ Ww2,
                                   const float* __restrict__ bw2,
                                   float* __restrict__ wscore, int NW) {
  int w = threadIdx.x;
  if (w >= 32) return;
  float s = 0.f;
  if (w < NW) {
    for (int j = 0; j < HID; ++j) {
      float t = bw1[j];
      for (int d = 0; d < 20; ++d) t += emb[w * 20 + d] * Ww1[d * HID + j];
      t = t > 0.f ? t : 0.f;
      s += t * Ww2[j];
    }
    s += bw2[0];
  }
  wscore[w] = s;
}

// ---- main fused scorer: score[n] = relu(x[n]@W1+b1)@W2 + b2 + wprior ------
// 256 threads = 8 waves; wave w owns candidate rows [w*16, w*16+16).
// A (16x4 f32): lanes 0-15 K={0,1}, lanes 16-31 K={2,3} (per-lane v2f).
// B (4x16 f32): mirrored K split; W1 chunk stored K-transposed in LDS so a
//               B fragment is one 8B ds_load_b64 of two consecutive K.
// X tile goes HBM->LDS via async-to-LDS loads (no VGPR staging); each wave
// drains its own ASYNCcnt before the block barrier.
__global__ __launch_bounds__(256)
void score_wmma_kernel(const float* __restrict__ x,
                       const int* __restrict__ widx,
                       const float* __restrict__ W1,
                       const float* __restrict__ b1,
                       const float* __restrict__ W2,
                       const float* __restrict__ b2,
                       const float* __restrict__ wscore,
                       float* __restrict__ out_scores,
                       unsigned* __restrict__ keys, int N) {
  __shared__ __align__(16) float xs[ROWS_PER_BLOCK * XS_STRIDE]; // 34.8 KB
  __shared__ __align__(16) float wt[HID * WT_STRIDE];            // 34.8 KB
  const int tid  = threadIdx.x;
  const int wave = tid >> 5;
  const int lane = tid & 31;
  const int l15  = lane & 15;
  const int lhi  = lane >> 4;
  const int row0 = blockIdx.x * ROWS_PER_BLOCK;

  v8f acc[8] = {}; // 8 column tiles of H=128, 16x16 f32 each

  for (int kc = 0; kc < D_SPAN; kc += KC) {
    __syncthreads(); // previous chunk's compute done before LDS overwrite
    { // stage X tile [128 x 64]: coalesced 16B transfers, HBM -> LDS
      const int r0 = tid >> 4;
      const int c  = (tid & 15) << 2;
      #pragma unroll
      for (int p = 0; p < 8; ++p) {
        int r = r0 + p * 16;
        int g = row0 + r; if (g >= N) g = N - 1;
#ifdef HAS_ASYNC_LDS
        float* gsrc = const_cast<float*>(x + (size_t)g * D_SPAN + kc + c);
        float* ldst = xs + r * XS_STRIDE + c;
        __builtin_amdgcn_global_load_async_to_lds_b128(
            (gv4i_ptr)gsrc, (lv4i_ptr)ldst, 0, 0);
#else
        float4 v = *(const float4*)(x + (size_t)g * D_SPAN + kc + c);
        *(float4*)(xs + r * XS_STRIDE + c) = v;
#endif
      }
    }
    { // stage W1 chunk transposed: wt[n][kr] = W1[kc+kr][n] (L2-resident)
      #pragma unroll
      for (int i = 0; i < (KC * HID / 256); ++i) {
        int idx = i * 256 + tid;
        int kr  = idx >> 7;
        int n   = idx & 127;
        wt[n * WT_STRIDE + kr] = W1[(size_t)(kc + kr) * HID + n];
      }
    }
#ifdef HAS_ASYNC_LDS
#ifdef HAS_WAIT_ASYNC_BUILTIN
    __builtin_amdgcn_s_wait_asynccnt(0);
#else
    asm volatile("s_wait_asynccnt 0" ::: "memory");
#endif
#endif
    __syncthreads();

    const float* xa = xs + (wave * 16 + l15) * XS_STRIDE + lhi * 2;
    #pragma unroll 4
    for (int kk = 0; kk < KC; kk += 4) {
      v2f a = *(const v2f*)(xa + kk);
      #pragma unroll
      for (int ct = 0; ct < 8; ++ct) {
        v2f b = *(const v2f*)(wt + (ct * 16 + l15) * WT_STRIDE + kk + lhi * 2);
        acc[ct] = __builtin_amdgcn_wmma_f32_16x16x4_f32(
            false, a, false, b, (short)0, acc[ct], false, false);
      }
    }
  }

  // epilogue: bias + ReLU + dot with W2, reduce over N (H columns).
  // C/D layout: lane holds N=l15 (+ct*16), rows M = r + lhi*8 for r=0..7.
  float part[8];
  #pragma unroll
  for (int r = 0; r < 8; ++r) part[r] = 0.f;
  #pragma unroll
  for (int ct = 0; ct < 8; ++ct) {
    int n = ct * 16 + l15;
    float bb = b1[n], w2 = W2[n];
    #pragma unroll
    for (int r = 0; r < 8; ++r) {
      float h = acc[ct][r] + bb;
      h = h > 0.f ? h : 0.f;
      part[r] += h * w2;
    }
  }
  #pragma unroll
  for (int off = 1; off < 16; off <<= 1) { // 16-lane butterfly (wave32 safe)
    #pragma unroll
    for (int r = 0; r < 8; ++r) part[r] += __shfl_xor(part[r], off, 32);
  }
  if (l15 < 8) {
    int m = row0 + wave * 16 + lhi * 8 + l15;
    if (m < N) {
      float s = part[l15] + b2[0] + wscore[widx[m]];
      out_scores[m] = s;
      keys[m] = mono_key(s);
    }
  }
}

// ---- radix-select pass: shared-memory histogram of a key bit-field --------
__global__ __launch_bounds__(256)
void hist_kernel(const unsigned* __restrict__ keys, int N,
                 unsigned* __restrict__ hist, const unsigned* __restrict__ scal,
                 int mode, int iters) {
  __shared__ unsigned sh[2048];
  int t = threadIdx.x;
  for (int i = t; i < 2048; i += 256) sh[i] = 0u;
  __syncthreads();
  unsigned p1 = scal[0], p21 = scal[2];
  int base = blockIdx.x * iters * 256;
  for (int i = 0; i < iters; ++i) {
    int n = base + i * 256 + t;
    if (n < N) {
      unsigned key = keys[n];
      if (mode == 0) atomicAdd(&sh[key >> 21], 1u);
      else if (mode == 1) { if ((key >> 21) == p1) atomicAdd(&sh[(key >> 10) & 2047u], 1u); }
      else               { if ((key >> 10) == p21) atomicAdd(&sh[key & 1023u], 1u); }
    }
  }
  __syncthreads();
  int bins = (mode == 2) ? 1024 : 2048;
  for (int i = t; i < bins; i += 256) {
    unsigned v = sh[i];
    if (v) atomicAdd(&hist[i], v);
  }
}

// ---- find the bin containing the k-th largest (scan from the top) ---------
__global__ __launch_bounds__(256)
void select_kernel(const unsigned* __restrict__ hist, unsigned* __restrict__ scal,
                   int mode, unsigned k_host) {
  __shared__ unsigned suf[256];
  int t = threadIdx.x;
  int bins  = (mode == 2) ? 1024 : 2048;
  int chunk = bins / 256;
  unsigned target = (mode == 0) ? k_host : ((mode == 1) ? scal[1] : scal[3]);
  unsigned s = 0;
  for (int j = 0; j < chunk; ++j) s += hist[t * chunk + j];
  suf[t] = s;
  __syncthreads();
  for (int off = 1; off < 256; off <<= 1) { // inclusive suffix sum
    unsigned v = (t + off < 256) ? suf[t + off] : 0u;
    __syncthreads();
    suf[t] += v;
    __syncthreads();
  }
  unsigned above = suf[t] - s; // count in chunks strictly above mine
  if (above < target && above + s >= target) { // unique thread
    unsigned run = above;
    for (int j = chunk - 1; j >= 0; --j) {
      unsigned c = hist[t * chunk + j];
      if (run + c >= target) {
        unsigned bin = (unsigned)(t * chunk + j);
        unsigned quota = target - run;
        if (mode == 0)      { scal[0] = bin; scal[1] = quota; }
        else if (mode == 1) { scal[2] = (scal[0] << 11) | bin; scal[3] = quota; }
        else                { scal[4] = (scal[2] << 10) | bin; scal[5] = quota; }
        break;
      }
      run += c;
    }
  }
}

// ---- per-block (gt, eq) counts vs exact threshold T -----------------------
__global__ __launch_bounds__(256)
void count_kernel(const unsigned* __restrict__ keys, int N,
                  const unsigned* __restrict__ scal, unsigned* __restrict__ bs,
                  int iters) {
  __shared__ unsigned sg[256], se[256];
  int t = threadIdx.x;
  unsigned T = scal[4];
  int base = blockIdx.x * iters * 256;
  unsigned g = 0, e = 0;
  for (int i = 0; i < iters; ++i) {
    int n = base + i * 256 + t;
    if (n < N) { unsigned key = keys[n]; g += (key > T); e += (key == T); }
  }
  sg[t] = g; se[t] = e;
  __syncthreads();
  for (int off = 128; off > 0; off >>= 1) {
    if (t < off) { sg[t] += sg[t + off]; se[t] += se[t + off]; }
    __syncthreads();
  }
  if (t == 0) { bs[2 * blockIdx.x] = sg[0]; bs[2 * blockIdx.x + 1] = se[0]; }
}

// ---- exclusive scan of 256 (gt,eq) pairs ----------------------------------
__global__ __launch_bounds__(256)
void scan_pairs_kernel(const unsigned* __restrict__ bs,
                       unsigned* __restrict__ boff, int NB) {
  __shared__ unsigned long long sc[256];
  int t = threadIdx.x;
  unsigned long long v = 0;
  if (t < NB) v = ((unsigned long long)bs[2 * t] << 32) | bs[2 * t + 1];
  sc[t] = v;
  __syncthreads();
  for (int off = 1; off < 256; off <<= 1) {
    unsigned long long u = (t >= off) ? sc[t - off] : 0ull;
    __syncthreads();
    sc[t] += u;
    __syncthreads();
  }
  unsigned long long excl = sc[t] - v;
  if (t < NB) {
    boff[2 * t]     = (unsigned)(excl >> 32);
    boff[2 * t + 1] = (unsigned)(excl & 0xffffffffull);
  }
}

// ---- emit: positions via prefix sums => top_idx sorted ascending ----------
__global__ __launch_bounds__(256)
void emit_kernel(const unsigned* __restrict__ keys, const int* __restrict__ beg,
                 const int* __restrict__ endp, const float* __restrict__ scores,
                 int N, const unsigned* __restrict__ scal,
                 const unsigned* __restrict__ boff, int iters, int k,
                 int* __restrict__ sel_idx, float* __restrict__ out, int outBase) {
  __shared__ unsigned sc[256];
  int t = threadIdx.x;
  unsigned T = scal[4], needEq = scal[5];
  int base = blockIdx.x * iters * 256;
  int nb = base + t * iters; // contiguous per-thread range keeps index order
  unsigned myg = 0, mye = 0;
  for (int i = 0; i < iters; ++i) {
    int n = nb + i;
    if (n < N) { unsigned key = keys[n]; myg += (key > T); mye += (key == T); }
  }
  unsigned pack = (myg << 16) | mye; // block totals <= 1024, fits 16 bits
  sc[t] = pack;
  __syncthreads();
  for (int off = 1; off < 256; off <<= 1) {
    unsigned v = (t >= off) ? sc[t - off] : 0u;
    __syncthreads();
    sc[t] += v;
    __syncthreads();
  }
  unsigned excl = sc[t] - pack;
  unsigned gtb = boff[2 * blockIdx.x]     + (excl >> 16);
  unsigned eqb = boff[2 * blockIdx.x + 1] + (excl & 0xffffu);
  for (int i = 0; i < iters; ++i) {
    int n = nb + i;
    if (n < N) {
      unsigned key = keys[n];
      bool isg = key > T, ise = key == T;
      if (isg || (ise && eqb < needEq)) {
        unsigned pos = gtb + (eqb < needEq ? eqb : needEq);
        if (pos < (unsigned)k) {
          sel_idx[pos]            = n;
          out[outBase + pos]          = scores[n];       // sel_scores
          out[outBase + k + pos]      = (float)beg[n];   // sel_begin
          out[outBase + 2 * k + pos]  = (float)endp[n];  // sel_end
          out[outBase + 3 * k + pos]  = (float)n;        // top_idx
        }
      }
      gtb += isg; eqb += ise;
    }
  }
}

// ---- gather selected rows: one block per row, 256 x float4 = 1024 floats --
__global__ __launch_bounds__(256)
void gather_kernel(const float* __restrict__ x, const int* __restrict__ sel_idx,
                   float* __restrict__ out_vecs, int k) {
  int row = blockIdx.x;
  if (row >= k) return;
  int idx = sel_idx[row];
  const float4* src = (const float4*)(x + (size_t)idx * D_SPAN);
  float4*       dst = (float4*)(out_vecs + (size_t)row * D_SPAN);
  dst[threadIdx.x] = src[threadIdx.x];
}

extern "C" void kernel_launch(void* const* d_in, const int* in_sizes, int n_in,
                              void* d_out, int out_size, void* d_ws, size_t ws_size,
                              hipStream_t stream) {
  const float* x    = (const float*)d_in[0];
  const int*   beg  = (const int*)d_in[1];
  const int*   endp = (const int*)d_in[2];
  const int*   widx = (const int*)d_in[3];
  const float* emb  = (const float*)d_in[4];
  const float* W1   = (const float*)d_in[5];
  const float* b1   = (const float*)d_in[6];
  const float* W2   = (const float*)d_in[7];
  const float* b2   = (const float*)d_in[8];
  const float* Ww1  = (const float*)d_in[9];
  const float* bw1  = (const float*)d_in[10];
  const float* Ww2  = (const float*)d_in[11];
  const float* bw2  = (const float*)d_in[12];

  const int N  = in_sizes[1];
  const int NW = in_sizes[4] / 20;
  const int k  = (out_size - N) / (D_SPAN + 4); // out = scores + k*(D+4)

  char* ws = (char*)d_ws;
  float*    wscore = (float*)(ws + WS_WIDTH);
  unsigned* scal   = (unsigned*)(ws + WS_SCAL);
  unsigned* h1     = (unsigned*)(ws + WS_HIST1);
  unsigned* h2     = (unsigned*)(ws + WS_HIST2);
  unsigned* h3     = (unsigned*)(ws + WS_HIST3);
  unsigned* bs     = (unsigned*)(ws + WS_BS);
  unsigned* boff   = (unsigned*)(ws + WS_BOFF);
  int*      sel    = (int*)(ws + WS_SELIDX);
  unsigned* keys   = (unsigned*)(ws + WS_KEYS);
  float*    out    = (float*)d_out;

  // zero scalars + all three histograms (contiguous region)
  const int zcount = (WS_HIST3 + 4096 - WS_SCAL) / 4;
  zero_kernel<<<21, 256, 0, stream>>>(scal, zcount);

  width_score_kernel<<<1, 32, 0, stream>>>(emb, Ww1, bw1, Ww2, bw2, wscore, NW);

  const int nblocks = (N + ROWS_PER_BLOCK - 1) / ROWS_PER_BLOCK;
  score_wmma_kernel<<<nblocks, 256, 0, stream>>>(x, widx, W1, b1, W2, b2,
                                                 wscore, out, keys, N);

  const int iters = (N + 256 * 256 - 1) / (256 * 256); // elems/thread, 256 blocks

  hist_kernel<<<256, 256, 0, stream>>>(keys, N, h1, scal, 0, iters);
  select_kernel<<<1, 256, 0, stream>>>(h1, scal, 0, (unsigned)k);
  hist_kernel<<<256, 256, 0, stream>>>(keys, N, h2, scal, 1, iters);
  select_kernel<<<1, 256, 0, stream>>>(h2, scal, 1, (unsigned)k);
  hist_kernel<<<256, 256, 0, stream>>>(keys, N, h3, scal, 2, iters);
  select_kernel<<<1, 256, 0, stream>>>(h3, scal, 2, (unsigned)k);

  count_kernel<<<256, 256, 0, stream>>>(keys, N, scal, bs, iters);
  scan_pairs_kernel<<<1, 256, 0, stream>>>(bs, boff, 256);
  emit_kernel<<<256, 256, 0, stream>>>(keys, beg, endp, out, N, scal, boff,
                                       iters, k, sel, out, N + k * D_SPAN);
  gather_kernel<<<k, 256, 0, stream>>>(x, sel, out + N, k);
}